// StaticAttentionBaseline_35296041239259
// MI455X (gfx1250) — compile-verified
//
#include <hip/hip_runtime.h>
#include <stdint.h>

// CDNA5 / gfx1250, wave32. V_WMMA_F32_16X16X4_F32 (fp32-exact) for all dense
// layers; async global->LDS (ASYNCcnt) double-buffered weight staging for the
// dominant GEMM; VALU for softmax / batched glimpse matvec.

typedef __attribute__((ext_vector_type(2))) float v2f;
typedef __attribute__((ext_vector_type(8))) float v8f;

__device__ __forceinline__ v8f wmma_f32_k4(v2f a, v2f b, v8f c) {
  // 8-arg pattern: (neg_a, A, neg_b, B, c_mod, C, reuse_a, reuse_b)
  return __builtin_amdgcn_wmma_f32_16x16x4_f32(false, a, false, b, (short)0, c,
                                               false, false);
}

__device__ __forceinline__ v8f v8zero() {
  v8f z;
#pragma unroll
  for (int i = 0; i < 8; ++i) z[i] = 0.0f;
  return z;
}

// Async copy 16B/lane from global to LDS (VFLAT async encoding, ASYNCcnt).
__device__ __forceinline__ void async_load_b128(uint32_t lds_addr,
                                                const void* gptr) {
  asm volatile("global_load_async_to_lds_b128 %0, %1, off"
               :
               : "v"(lds_addr), "v"((uint64_t)(uintptr_t)gptr)
               : "memory");
}
__device__ __forceinline__ void wait_async0() {
  asm volatile("s_wait_asynccnt 0x0" ::: "memory");
}

// ---------------------------------------------------------------------------
// Kernel 1: visible_emb = relu(visible @ ve_w1 + b1) @ ve_w2 + b2
// rows = B*64 (row = b*64 + c). emb laid out as [B][1024] == flat_scene.
// ---------------------------------------------------------------------------
__global__ void __launch_bounds__(256)
k_visible_emb(const float* __restrict__ scene,
              const float* __restrict__ w1, const float* __restrict__ b1,
              const float* __restrict__ w2, const float* __restrict__ b2,
              float* __restrict__ emb, int nrows) {
  __shared__ float lds_h1[8][256];
  const int lane = threadIdx.x & 31;
  const int wv   = threadIdx.x >> 5;
  const int tile = blockIdx.x * 8 + wv;
  if (tile * 16 >= nrows) return;  // wave-uniform
  const int row0 = tile * 16;
  const int hr   = lane & 15;          // A-row / B-col / D-col within tile
  const int kh   = (lane >> 4) * 2;    // K sub-offset for A/B frags
  const int rofs = 8 * (lane >> 4);    // D-row offset
  const long arow = row0 + hr;

  // GEMM1: A = visible rows (stride 22, 8 cols), B = ve_w1 (8x16)
  v8f c1 = v8zero();
#pragma unroll
  for (int k0 = 0; k0 < 8; k0 += 4) {
    v2f a, b;
    a.x = scene[arow * 22 + (k0 + kh)];
    a.y = scene[arow * 22 + (k0 + kh + 1)];
    b.x = w1[(k0 + kh) * 16 + hr];
    b.y = w1[(k0 + kh + 1) * 16 + hr];
    c1 = wmma_f32_k4(a, b, c1);
  }
  const float bias1 = b1[hr];
  float* h1 = lds_h1[wv];
#pragma unroll
  for (int r = 0; r < 8; ++r) {
    float v = c1[r] + bias1;
    h1[(r + rofs) * 16 + hr] = v > 0.0f ? v : 0.0f;  // relu, D->LDS
  }
  __syncthreads();

  // GEMM2: A = h1 (16x16 from LDS), B = ve_w2 (16x16)
  v8f c2 = v8zero();
#pragma unroll
  for (int k0 = 0; k0 < 16; k0 += 4) {
    v2f a, b;
    a.x = h1[hr * 16 + (k0 + kh)];
    a.y = h1[hr * 16 + (k0 + kh + 1)];
    b.x = w2[(k0 + kh) * 16 + hr];
    b.y = w2[(k0 + kh + 1) * 16 + hr];
    c2 = wmma_f32_k4(a, b, c2);
  }
  const float bias2 = b2[hr];
#pragma unroll
  for (int r = 0; r < 8; ++r)
    emb[(long)(row0 + r + rofs) * 16 + hr] = c2[r] + bias2;
}

// ---------------------------------------------------------------------------
// Kernel 2: state = relu([emb | cue_emb] @ ss_w1 + b1) @ ss_w2 + b2
// M = 16 batches per wave, K = 1032, N = 64 then 32.
// ss_w1 K-panels (64x64 = 16KB) are async-staged into LDS, double-buffered,
// shared by all 8 waves of the workgroup (8x less global weight traffic).
// ---------------------------------------------------------------------------
__global__ void __launch_bounds__(256)
k_state(const float* __restrict__ emb, const int* __restrict__ cue,
        const float* __restrict__ cue_tab,
        const float* __restrict__ w1, const float* __restrict__ b1,
        const float* __restrict__ w2, const float* __restrict__ b2,
        float* __restrict__ state, int B) {
  __shared__ __align__(16) float wbuf[2][64 * 64];  // 2 x 16KB weight panels
  __shared__ __align__(16) float wtail[8 * 64];     // ss_w1 rows 1024..1031
  __shared__ float lds_h[8][16 * 64];
  const int tid  = threadIdx.x;
  const int lane = tid & 31;
  const int wv   = tid >> 5;
  int tile = blockIdx.x * 8 + wv;
  const int maxt = B / 16 - 1;
  if (tile > maxt) tile = maxt;  // clamp: duplicate identical work is benign,
                                 // keeps all waves in cooperative barriers
  const int b0 = tile * 16;
  const int hr = lane & 15;
  const int kh = (lane >> 4) * 2;
  const int rofs = 8 * (lane >> 4);
  const long brow = b0 + hr;
  const int cueb = cue[brow];

  // stage tail rows (8x64 floats) with plain loads
  for (int i = tid; i < 512; i += 256) wtail[i] = w1[1024 * 64 + i];

  // async staging: one 64x64 K-panel = 4096 floats; 4 x B128 per thread
  auto stage = [&](int kc, int buf) {
#pragma unroll
    for (int j = 0; j < 4; ++j) {
      const int f4 = tid + j * 256;  // float4 index within the panel
      async_load_b128((uint32_t)(uintptr_t)&wbuf[buf][f4 * 4],
                      w1 + (size_t)kc * 64 + (size_t)f4 * 4);
    }
  };

  stage(0, 0);
  wait_async0();
  __syncthreads();

  v8f acc[4];
#pragma unroll
  for (int nt = 0; nt < 4; ++nt) acc[nt] = v8zero();

  const float* arow = emb + brow * 1024;
  for (int c = 0; c < 16; ++c) {  // 16 panels x 64 K-values
    const int kc = c * 64;
    if (c + 1 < 16) stage(kc + 64, (c + 1) & 1);  // prefetch next panel
    const float* wb = wbuf[c & 1];
#pragma unroll
    for (int kk = 0; kk < 64; kk += 4) {
      v2f a;
      a.x = arow[kc + kk + kh];
      a.y = arow[kc + kk + kh + 1];
#pragma unroll
      for (int nt = 0; nt < 4; ++nt) {
        v2f b;
        b.x = wb[(kk + kh) * 64 + nt * 16 + hr];
        b.y = wb[(kk + kh + 1) * 64 + nt * 16 + hr];
        acc[nt] = wmma_f32_k4(a, b, acc[nt]);
      }
    }
    wait_async0();   // next panel resident
    __syncthreads(); // everyone done reading current panel
  }

  // cue-embedding tail of K (rows 1024..1031)
#pragma unroll
  for (int k0 = 0; k0 < 8; k0 += 4) {
    v2f a;
    a.x = cue_tab[cueb * 8 + k0 + kh];
    a.y = cue_tab[cueb * 8 + k0 + kh + 1];
#pragma unroll
    for (int nt = 0; nt < 4; ++nt) {
      v2f b;
      b.x = wtail[(k0 + kh) * 64 + nt * 16 + hr];
      b.y = wtail[(k0 + kh + 1) * 64 + nt * 16 + hr];
      acc[nt] = wmma_f32_k4(a, b, acc[nt]);
    }
  }

  float* hl = lds_h[wv];
#pragma unroll
  for (int nt = 0; nt < 4; ++nt) {
    const float bv = b1[nt * 16 + hr];
#pragma unroll
    for (int r = 0; r < 8; ++r) {
      float v = acc[nt][r] + bv;
      hl[(r + rofs) * 64 + nt * 16 + hr] = v > 0.0f ? v : 0.0f;
    }
  }
  __syncthreads();

  v8f st[2];
  st[0] = v8zero();
  st[1] = v8zero();
#pragma unroll
  for (int k0 = 0; k0 < 64; k0 += 4) {
    v2f a;
    a.x = hl[hr * 64 + k0 + kh];
    a.y = hl[hr * 64 + k0 + kh + 1];
#pragma unroll
    for (int nt = 0; nt < 2; ++nt) {
      v2f b;
      b.x = w2[(k0 + kh) * 32 + nt * 16 + hr];
      b.y = w2[(k0 + kh + 1) * 32 + nt * 16 + hr];
      st[nt] = wmma_f32_k4(a, b, st[nt]);
    }
  }
#pragma unroll
  for (int nt = 0; nt < 2; ++nt) {
    const float bv = b2[nt * 16 + hr];
#pragma unroll
    for (int r = 0; r < 8; ++r)
      state[(long)(b0 + r + rofs) * 32 + nt * 16 + hr] = st[nt][r] + bv;
  }
}

// ---------------------------------------------------------------------------
// Kernel 3: attention softmax, glimpse, obs, th-MLP, confidence, all outputs
// (replicated xS since every step is identical).
// ---------------------------------------------------------------------------
__global__ void __launch_bounds__(256)
k_tail(const float* __restrict__ scene, const int* __restrict__ cue,
       const float* __restrict__ state,
       const float* __restrict__ aw, const float* __restrict__ ab,
       const float* __restrict__ tw1, const float* __restrict__ tb1,
       const float* __restrict__ tw2, const float* __restrict__ tb2,
       float* __restrict__ o_last, float* __restrict__ o_logits,
       float* __restrict__ o_attn, float* __restrict__ o_obs,
       float* __restrict__ o_conf, float* __restrict__ o_loss,
       float* __restrict__ o_state, int B, int S) {
  __shared__ float s_attn[8][1024];
  __shared__ float s_x[8][16 * 44];
  __shared__ float s_h[8][1024];
  __shared__ float s_gl[8][16 * 14];
  __shared__ float s_lg[8][256];
  __shared__ float s_cf[8][16];
  const int lane = threadIdx.x & 31;
  const int wv   = threadIdx.x >> 5;
  const int tile = blockIdx.x * 8 + wv;
  if (tile * 16 >= B) return;
  const int b0 = tile * 16;
  const int hr = lane & 15;
  const int kh = (lane >> 4) * 2;
  const int rofs = 8 * (lane >> 4);

  // --- attn logits: state(16x32) @ attn_w(32,64), scaled by 1/TEMP
  v8f acc[4];
#pragma unroll
  for (int nt = 0; nt < 4; ++nt) acc[nt] = v8zero();
#pragma unroll
  for (int k0 = 0; k0 < 32; k0 += 4) {
    v2f a;
    a.x = state[(long)(b0 + hr) * 32 + k0 + kh];
    a.y = state[(long)(b0 + hr) * 32 + k0 + kh + 1];
#pragma unroll
    for (int nt = 0; nt < 4; ++nt) {
      v2f b;
      b.x = aw[(k0 + kh) * 64 + nt * 16 + hr];
      b.y = aw[(k0 + kh + 1) * 64 + nt * 16 + hr];
      acc[nt] = wmma_f32_k4(a, b, acc[nt]);
    }
  }
  const float invT = 1.0f / 0.6f;
  float* at = s_attn[wv];
#pragma unroll
  for (int nt = 0; nt < 4; ++nt) {
    const float bv = ab[nt * 16 + hr];
#pragma unroll
    for (int r = 0; r < 8; ++r)
      at[(r + rofs) * 64 + nt * 16 + hr] = (acc[nt][r] + bv) * invT;
  }
  __syncthreads();

  // --- softmax over 64 channels (lanes 0..15, one batch each)
  if (lane < 16) {
    float m = -1e30f;
    for (int c = 0; c < 64; ++c) m = fmaxf(m, at[lane * 64 + c]);
    float s = 0.0f;
    for (int c = 0; c < 64; ++c) {
      float e = __expf(at[lane * 64 + c] - m);
      at[lane * 64 + c] = e;
      s += e;
    }
    const float rs = 1.0f / s;
    for (int c = 0; c < 64; ++c) at[lane * 64 + c] *= rs;
  }
  __syncthreads();

  // --- glimpse[b][h] = sum_c attn[b][c] * hidden[b][c][h], h split over halves
  {
    const int b = hr;
    const int hbase = (lane >> 4) * 7;
    const float* srow = scene + (long)(b0 + b) * 1408 + 8;  // hidden offset
    float* gl = s_gl[wv];
    for (int hh = 0; hh < 7; ++hh) {
      const int h = hbase + hh;
      float g = 0.0f;
      for (int c = 0; c < 64; ++c) g += at[b * 64 + c] * srow[c * 22 + h];
      gl[b * 14 + h] = g;
    }
  }
  __syncthreads();

  // --- obs + build x = [obs(11) | state(32) | 0pad] (K padded to 44)
  float* xr = s_x[wv];
  if (lane < 16) {
    const int b = lane;
    const int cb = cue[b0 + b];
    const float* gl = s_gl[wv] + b * 14;
    xr[b * 44 + 0] = gl[cb];  // matched = glimpse[b][cue[b]] (one-hot dot)
    for (int j = 0; j < 10; ++j) xr[b * 44 + 1 + j] = gl[4 + j];
    for (int j = 0; j < 32; ++j) xr[b * 44 + 11 + j] = state[(long)(b0 + b) * 32 + j];
    xr[b * 44 + 43] = 0.0f;
  }
  __syncthreads();

  // --- th GEMM1: (16x44) @ th_w1(43x64), rows >= 43 are zero
#pragma unroll
  for (int nt = 0; nt < 4; ++nt) acc[nt] = v8zero();
#pragma unroll
  for (int k0 = 0; k0 < 44; k0 += 4) {
    const int ka = k0 + kh;
    v2f a;
    a.x = xr[hr * 44 + ka];
    a.y = xr[hr * 44 + ka + 1];
#pragma unroll
    for (int nt = 0; nt < 4; ++nt) {
      v2f b;
      b.x = (ka < 43) ? tw1[ka * 64 + nt * 16 + hr] : 0.0f;
      b.y = (ka + 1 < 43) ? tw1[(ka + 1) * 64 + nt * 16 + hr] : 0.0f;
      acc[nt] = wmma_f32_k4(a, b, acc[nt]);
    }
  }
  float* hb = s_h[wv];
#pragma unroll
  for (int nt = 0; nt < 4; ++nt) {
    const float bv = tb1[nt * 16 + hr];
#pragma unroll
    for (int r = 0; r < 8; ++r) {
      float v = acc[nt][r] + bv;
      hb[(r + rofs) * 64 + nt * 16 + hr] = v > 0.0f ? v : 0.0f;
    }
  }
  __syncthreads();

  // --- th GEMM2: (16x64) @ th_w2(64x10); cols >= 10 are zero
  v8f lg = v8zero();
#pragma unroll
  for (int k0 = 0; k0 < 64; k0 += 4) {
    v2f a;
    a.x = hb[hr * 64 + k0 + kh];
    a.y = hb[hr * 64 + k0 + kh + 1];
    v2f b;
    b.x = (hr < 10) ? tw2[(k0 + kh) * 10 + hr] : 0.0f;
    b.y = (hr < 10) ? tw2[(k0 + kh + 1) * 10 + hr] : 0.0f;
    lg = wmma_f32_k4(a, b, lg);
  }
  const float bv2 = (hr < 10) ? tb2[hr] : 0.0f;
  float* lgl = s_lg[wv];
#pragma unroll
  for (int r = 0; r < 8; ++r) lgl[(r + rofs) * 16 + hr] = lg[r] + bv2;
  __syncthreads();

  // --- confidence = max softmax prob = 1 / sum(exp(l - max))
  if (lane < 16) {
    const int b = lane;
    float m = -1e30f;
    for (int j = 0; j < 10; ++j) m = fmaxf(m, lgl[b * 16 + j]);
    float s = 0.0f;
    for (int j = 0; j < 10; ++j) s += __expf(lgl[b * 16 + j] - m);
    s_cf[wv][b] = 1.0f / s;
  }
  __syncthreads();

  // --- outputs (step axis is a pure replication)
  for (int i = lane; i < 16 * S * 64; i += 32) {
    int b = i / (S * 64), rem = i % (S * 64), s = rem / 64, c = rem % 64;
    o_attn[((long)(b0 + b) * S + s) * 64 + c] = at[b * 64 + c];
  }
  for (int i = lane; i < 16 * S * 10; i += 32) {
    int b = i / (S * 10), rem = i % (S * 10), s = rem / 10, j = rem % 10;
    o_logits[((long)(b0 + b) * S + s) * 10 + j] = lgl[b * 16 + j];
  }
  for (int i = lane; i < 160; i += 32) {
    int b = i / 10, j = i % 10;
    o_last[(long)(b0 + b) * 10 + j] = lgl[b * 16 + j];
  }
  for (int i = lane; i < 16 * S * 11; i += 32) {
    int b = i / (S * 11), rem = i % (S * 11), s = rem / 11, j = rem % 11;
    o_obs[((long)(b0 + b) * S + s) * 11 + j] = xr[b * 44 + j];
  }
  for (int i = lane; i < 16 * S; i += 32) {
    int b = i / S, s = i % S;
    float cf = s_cf[wv][b];
    o_conf[(long)(b0 + b) * S + s] = cf;
    o_loss[(long)(b0 + b) * S + s] = -cf;
  }
  for (int i = lane; i < 16 * S * 32; i += 32) {
    int b = i / (S * 32), rem = i % (S * 32), s = rem / 32, j = rem % 32;
    o_state[((long)(b0 + b) * S + s) * 32 + j] = state[(long)(b0 + b) * 32 + j];
  }
}

// ---------------------------------------------------------------------------
extern "C" void kernel_launch(void* const* d_in, const int* in_sizes, int n_in,
                              void* d_out, int out_size, void* d_ws,
                              size_t ws_size, hipStream_t stream) {
  const float* scene   = (const float*)d_in[0];
  const int*   cue     = (const int*)d_in[1];
  const float* cue_tab = (const float*)d_in[2];
  const float* ve_w1   = (const float*)d_in[3];
  const float* ve_b1   = (const float*)d_in[4];
  const float* ve_w2   = (const float*)d_in[5];
  const float* ve_b2   = (const float*)d_in[6];
  const float* ss_w1   = (const float*)d_in[7];
  const float* ss_b1   = (const float*)d_in[8];
  const float* ss_w2   = (const float*)d_in[9];
  const float* ss_b2   = (const float*)d_in[10];
  const float* th_w1   = (const float*)d_in[11];
  const float* th_b1   = (const float*)d_in[12];
  const float* th_w2   = (const float*)d_in[13];
  const float* th_b2   = (const float*)d_in[14];
  const float* attn_w  = (const float*)d_in[15];
  const float* attn_b  = (const float*)d_in[16];

  const int B = in_sizes[1];                                // cue is (B,)
  const int S = (B > 0) ? ((out_size / B - 10) / 119) : 8;  // per-B = 10+119S

  float* emb   = (float*)d_ws;             // [B][1024]
  float* state = emb + (size_t)B * 1024;   // [B][32]

  float* out      = (float*)d_out;
  float* o_last   = out;                            // B*10
  float* o_logits = o_last + (size_t)B * 10;        // B*S*10
  float* o_attn   = o_logits + (size_t)B * S * 10;  // B*S*64
  float* o_obs    = o_attn + (size_t)B * S * 64;    // B*S*11
  float* o_conf   = o_obs + (size_t)B * S * 11;     // B*S
  float* o_loss   = o_conf + (size_t)B * S;         // B*S
  float* o_state  = o_loss + (size_t)B * S;         // B*S*32

  dim3 blk(256);
  const int nrows  = B * 64;
  const int tiles1 = nrows / 16;
  const int tiles2 = B / 16;

  k_visible_emb<<<(tiles1 + 7) / 8, blk, 0, stream>>>(scene, ve_w1, ve_b1,
                                                      ve_w2, ve_b2, emb, nrows);
  k_state<<<(tiles2 + 7) / 8, blk, 0, stream>>>(emb, cue, cue_tab, ss_w1,
                                                ss_b1, ss_w2, ss_b2, state, B);
  k_tail<<<(tiles2 + 7) / 8, blk, 0, stream>>>(
      scene, cue, state, attn_w, attn_b, th_w1, th_b1, th_w2, th_b2, o_last,
      o_logits, o_attn, o_obs, o_conf, o_loss, o_state, B, S);
}